// MoE_67997922231059
// MI455X (gfx1250) — compile-verified
//
#include <hip/hip_runtime.h>
#include <math.h>

// ---------------------------------------------------------------------------
// MoE top-2 FFN for gfx1250 (CDNA5), fp32 WMMA (V_WMMA_F32_16X16X4_F32).
// T=2048 tokens, d=768, d_ff=3072, E=8 experts, top-2 routing.
// GEMM tiles: M=64 x N=256, 8 waves (2x4), each wave 2x4 16x16 accumulators.
// A LDS stride 34 (even, conflict-free b64 frags); B LDS K-pair interleaved,
// row stride 544 (== 32 mod 64 banks -> lane-halves on disjoint bank halves).
// ---------------------------------------------------------------------------

#define T_TOK   2048
#define N_EMBD  768
#define D_FF    3072
#define EXPERTS 8
#define KC      32            // K chunk staged through LDS
#define LDA     34            // padded A row stride (floats)
#define NT      256           // N tile
#define SBS     544           // B LDS floats per K-pair row (512 data + 32 pad)

typedef __attribute__((ext_vector_type(2))) float v2f;
typedef __attribute__((ext_vector_type(8))) float v8f;

__device__ __forceinline__ v8f wmma_f32_16x16x4(v2f a, v2f b, v8f c) {
    // 8 args: (neg_a, A, neg_b, B, c_mod, C, reuse_a, reuse_b)
    return __builtin_amdgcn_wmma_f32_16x16x4_f32(false, a, false, b,
                                                 (short)0, c, false, false);
}

// ---------------------------------------------------------------------------
// K0: zero the per-expert counters
// ---------------------------------------------------------------------------
__global__ void moe_zero_counts(int* counts) {
    if (threadIdx.x < EXPERTS) counts[threadIdx.x] = 0;
}

// ---------------------------------------------------------------------------
// K1: router. One wave (32 lanes) per token.
// ---------------------------------------------------------------------------
__global__ __launch_bounds__(256) void moe_router(
    const float* __restrict__ X,       // [T, d]
    const float* __restrict__ NOISE,   // [T, E]
    const float* __restrict__ RW,      // [E, d]
    int*   __restrict__ counts,        // [E]
    int*   __restrict__ tok_idx,       // [E, T]
    float* __restrict__ tok_w,         // [E, T]
    int*   __restrict__ route_e,       // [T, 2]
    int*   __restrict__ route_slot)    // [T, 2]
{
    const int t    = blockIdx.x * 8 + (threadIdx.x >> 5);
    const int lane = threadIdx.x & 31;

    float acc[EXPERTS];
#pragma unroll
    for (int e = 0; e < EXPERTS; e++) acc[e] = 0.0f;

    const float* xr = X + (size_t)t * N_EMBD;
    for (int i = lane; i < N_EMBD; i += 32) {
        const float xv = xr[i];
#pragma unroll
        for (int e = 0; e < EXPERTS; e++)
            acc[e] += xv * RW[e * N_EMBD + i];
    }
    // wave32 tree reduction
#pragma unroll
    for (int off = 16; off >= 1; off >>= 1) {
#pragma unroll
        for (int e = 0; e < EXPERTS; e++)
            acc[e] += __shfl_xor(acc[e], off, 32);
    }

    if (lane == 0) {
        float lg[EXPERTS];
#pragma unroll
        for (int e = 0; e < EXPERTS; e++)
            lg[e] = acc[e] + NOISE[t * EXPERTS + e];

        // top-2 (lowest index wins ties, matching lax.top_k)
        int i0 = 0;
#pragma unroll
        for (int e = 1; e < EXPERTS; e++) if (lg[e] > lg[i0]) i0 = e;
        int i1 = (i0 == 0) ? 1 : 0;
#pragma unroll
        for (int e = 0; e < EXPERTS; e++)
            if (e != i0 && lg[e] > lg[i1]) i1 = e;

        // softmax over the two kept logits
        const float e1 = __expf(lg[i1] - lg[i0]);
        const float inv = 1.0f / (1.0f + e1);
        const float w0 = inv;
        const float w1 = e1 * inv;

        int s0 = atomicAdd(&counts[i0], 1);
        tok_idx[i0 * T_TOK + s0] = t;
        tok_w  [i0 * T_TOK + s0] = w0;
        route_e   [t * 2 + 0] = i0;
        route_slot[t * 2 + 0] = s0;

        int s1 = atomicAdd(&counts[i1], 1);
        tok_idx[i1 * T_TOK + s1] = t;
        tok_w  [i1 * T_TOK + s1] = w1;
        route_e   [t * 2 + 1] = i1;
        route_slot[t * 2 + 1] = s1;
    }
}

// ---------------------------------------------------------------------------
// K2: exclusive scan of 8 counts
// ---------------------------------------------------------------------------
__global__ void moe_scan(const int* __restrict__ counts, int* __restrict__ offsets) {
    if (threadIdx.x == 0) {
        int s = 0;
        for (int e = 0; e < EXPERTS; e++) { offsets[e] = s; s += counts[e]; }
        offsets[EXPERTS] = s;
    }
}

// ---------------------------------------------------------------------------
// shared GEMM helpers
// ---------------------------------------------------------------------------
__device__ __forceinline__ void stage_B(float* __restrict__ sB,
                                        const float* __restrict__ wsrc, // row k0
                                        int ld, int n0, int tid) {
    // KC x NT slice, stored K-pair interleaved: (k,n) -> sB[(k>>1)*SBS + n*2 + (k&1)]
    for (int i = tid; i < (KC / 2) * (NT / 4); i += 256) {
        const int kp = i >> 6;            // / (NT/4)
        const int n4 = (i & 63) << 2;
        const float* src = wsrc + (size_t)(2 * kp) * ld + n0 + n4;
        const float4 r0 = *(const float4*)(src);
        const float4 r1 = *(const float4*)(src + ld);
        float* dst = sB + kp * SBS + n4 * 2;
        v2f t;
        t.x = r0.x; t.y = r1.x; ((v2f*)dst)[0] = t;
        t.x = r0.y; t.y = r1.y; ((v2f*)dst)[1] = t;
        t.x = r0.z; t.y = r1.z; ((v2f*)dst)[2] = t;
        t.x = r0.w; t.y = r1.w; ((v2f*)dst)[3] = t;
    }
}

__device__ __forceinline__ void compute_chunk(const float* __restrict__ sA,
                                              const float* __restrict__ sB,
                                              v8f acc[2][4],
                                              int wm, int wn, int lane) {
    const int lm = lane & 15;
    const int lk = (lane >> 4) * 2;       // 0 or 2
#pragma unroll
    for (int k4 = 0; k4 < KC; k4 += 4) {
        v2f afr[2], bfr[4];
#pragma unroll
        for (int i = 0; i < 2; i++)
            afr[i] = *(const v2f*)(sA + (wm + i * 16 + lm) * LDA + k4 + lk);
        const int kp = (k4 >> 1) + (lane >> 4);
#pragma unroll
        for (int j = 0; j < 4; j++)
            bfr[j] = *(const v2f*)(sB + kp * SBS + (wn + j * 16 + lm) * 2);
#pragma unroll
        for (int i = 0; i < 2; i++)
#pragma unroll
            for (int j = 0; j < 4; j++)
                acc[i][j] = wmma_f32_16x16x4(afr[i], bfr[j], acc[i][j]);
    }
}

// ---------------------------------------------------------------------------
// K3: GEMM1 — H[base+row, :] = gelu( X[tok] @ W1[e] + b1[e] )
// ---------------------------------------------------------------------------
__global__ __launch_bounds__(256) void moe_ffn1(
    const float* __restrict__ X,        // [T, d]
    const float* __restrict__ W1,       // [E, d, d_ff]
    const float* __restrict__ B1,       // [E, d_ff]
    const int*   __restrict__ counts,
    const int*   __restrict__ offsets,
    const int*   __restrict__ tok_idx,  // [E, T]
    float*       __restrict__ H)        // [2T, d_ff] compact
{
    const int e     = blockIdx.z;
    const int count = counts[e];
    const int m0    = blockIdx.y * 64;
    if (m0 >= count) return;
    const int n0 = blockIdx.x * NT;

    __shared__ float sA[64 * LDA];
    __shared__ float sB[(KC / 2) * SBS];
    __shared__ int   sTok[64];

    const int tid  = threadIdx.x;
    const int lane = tid & 31;
    const int wid  = tid >> 5;
    const int wm   = (wid >> 2) * 32;    // wave M offset (0/32)
    const int wn   = (wid & 3) * 64;     // wave N offset (0/64/128/192)

    if (tid < 64) {
        const int r = m0 + tid;
        sTok[tid] = (r < count) ? tok_idx[e * T_TOK + r] : -1;
    }

    v8f acc[2][4];
#pragma unroll
    for (int i = 0; i < 2; i++)
#pragma unroll
        for (int j = 0; j < 4; j++) acc[i][j] = {};

    const float* w1e = W1 + (size_t)e * N_EMBD * D_FF;

    for (int k0 = 0; k0 < N_EMBD; k0 += KC) {
        __syncthreads();
        // A: 64 x KC gathered token rows
        for (int i = tid; i < 64 * (KC / 4); i += 256) {
            const int r  = i >> 3;
            const int c4 = (i & 7) << 2;
            const int trow = sTok[r];
            float4 v = make_float4(0.f, 0.f, 0.f, 0.f);
            if (trow >= 0)
                v = *(const float4*)(X + (size_t)trow * N_EMBD + k0 + c4);
            float* dst = sA + r * LDA + c4;    // 8B aligned (LDA even, c4 even)
            v2f t;
            t.x = v.x; t.y = v.y; ((v2f*)dst)[0] = t;
            t.x = v.z; t.y = v.w; ((v2f*)dst)[1] = t;
        }
        stage_B(sB, w1e + (size_t)k0 * D_FF, D_FF, n0, tid);
        __syncthreads();
        compute_chunk(sA, sB, acc, wm, wn, lane);
    }

    // epilogue: bias + gelu(tanh approx) -> H
    const int base = offsets[e];
    const float* b1e = B1 + e * D_FF;
    const int lm = lane & 15;
    const int lr = (lane >> 4) * 8;
#pragma unroll
    for (int i = 0; i < 2; i++) {
#pragma unroll
        for (int j = 0; j < 4; j++) {
            const int nn = n0 + wn + j * 16 + lm;
            const float bias = b1e[nn];
#pragma unroll
            for (int r = 0; r < 8; r++) {
                const int row = m0 + wm + i * 16 + lr + r;
                if (row < count) {
                    const float v = acc[i][j][r] + bias;
                    const float u = 0.7978845608028654f * (v + 0.044715f * v * v * v);
                    const float g = 0.5f * v * (1.0f + tanhf(u));
                    H[(size_t)(base + row) * D_FF + nn] = g;
                }
            }
        }
    }
}

// ---------------------------------------------------------------------------
// K4: GEMM2 — Y[base+row, :] = w_row * ( H[base+row, :] @ W2[e] + b2[e] )
// ---------------------------------------------------------------------------
__global__ __launch_bounds__(256) void moe_ffn2(
    const float* __restrict__ H,        // [2T, d_ff]
    const float* __restrict__ W2,       // [E, d_ff, d]
    const float* __restrict__ B2,       // [E, d]
    const int*   __restrict__ counts,
    const int*   __restrict__ offsets,
    const float* __restrict__ tok_w,    // [E, T]
    float*       __restrict__ Y)        // [2T, d]
{
    const int e     = blockIdx.z;
    const int count = counts[e];
    const int m0    = blockIdx.y * 64;
    if (m0 >= count) return;
    const int n0   = blockIdx.x * NT;
    const int base = offsets[e];

    __shared__ float sA[64 * LDA];
    __shared__ float sB[(KC / 2) * SBS];
    __shared__ float sW[64];

    const int tid  = threadIdx.x;
    const int lane = tid & 31;
    const int wid  = tid >> 5;
    const int wm   = (wid >> 2) * 32;
    const int wn   = (wid & 3) * 64;

    if (tid < 64) {
        const int r = m0 + tid;
        sW[tid] = (r < count) ? tok_w[e * T_TOK + r] : 0.0f;
    }

    v8f acc[2][4];
#pragma unroll
    for (int i = 0; i < 2; i++)
#pragma unroll
        for (int j = 0; j < 4; j++) acc[i][j] = {};

    const float* w2e = W2 + (size_t)e * D_FF * N_EMBD;

    for (int k0 = 0; k0 < D_FF; k0 += KC) {
        __syncthreads();
        for (int i = tid; i < 64 * (KC / 4); i += 256) {
            const int r  = i >> 3;
            const int c4 = (i & 7) << 2;
            const int hrow = m0 + r;
            float4 v = make_float4(0.f, 0.f, 0.f, 0.f);
            if (hrow < count)
                v = *(const float4*)(H + (size_t)(base + hrow) * D_FF + k0 + c4);
            float* dst = sA + r * LDA + c4;
            v2f t;
            t.x = v.x; t.y = v.y; ((v2f*)dst)[0] = t;
            t.x = v.z; t.y = v.w; ((v2f*)dst)[1] = t;
        }
        stage_B(sB, w2e + (size_t)k0 * N_EMBD, N_EMBD, n0, tid);
        __syncthreads();
        compute_chunk(sA, sB, acc, wm, wn, lane);
    }

    const float* b2e = B2 + e * N_EMBD;
    const int lm = lane & 15;
    const int lr = (lane >> 4) * 8;
#pragma unroll
    for (int i = 0; i < 2; i++) {
#pragma unroll
        for (int j = 0; j < 4; j++) {
            const int nn = n0 + wn + j * 16 + lm;
            const float bias = b2e[nn];
#pragma unroll
            for (int r = 0; r < 8; r++) {
                const int mloc = wm + i * 16 + lr + r;
                const int row  = m0 + mloc;
                if (row < count) {
                    const float v = (acc[i][j][r] + bias) * sW[mloc];
                    Y[(size_t)(base + row) * N_EMBD + nn] = v;
                }
            }
        }
    }
}

// ---------------------------------------------------------------------------
// K5: combine — out[t] = Y[row(t,0)] + Y[row(t,1)]
// ---------------------------------------------------------------------------
__global__ __launch_bounds__(192) void moe_combine(
    const float* __restrict__ Y,
    const int*   __restrict__ offsets,
    const int*   __restrict__ route_e,
    const int*   __restrict__ route_slot,
    float*       __restrict__ out)
{
    const int t  = blockIdx.x;
    const int e0 = route_e[t * 2 + 0];
    const int e1 = route_e[t * 2 + 1];
    const int r0 = offsets[e0] + route_slot[t * 2 + 0];
    const int r1 = offsets[e1] + route_slot[t * 2 + 1];
    const int n  = threadIdx.x * 4;
    const float4 a = *(const float4*)(Y + (size_t)r0 * N_EMBD + n);
    const float4 b = *(const float4*)(Y + (size_t)r1 * N_EMBD + n);
    float4 o;
    o.x = a.x + b.x; o.y = a.y + b.y; o.z = a.z + b.z; o.w = a.w + b.w;
    *(float4*)(out + (size_t)t * N_EMBD + n) = o;
}

// ---------------------------------------------------------------------------
extern "C" void kernel_launch(void* const* d_in, const int* in_sizes, int n_in,
                              void* d_out, int out_size, void* d_ws, size_t ws_size,
                              hipStream_t stream) {
    const float* X     = (const float*)d_in[0];   // [B,S,768]
    const float* NOISE = (const float*)d_in[1];   // [T,8]
    const float* RW    = (const float*)d_in[2];   // [8,768]
    const float* W1    = (const float*)d_in[3];   // [8,768,3072]
    const float* B1    = (const float*)d_in[4];   // [8,3072]
    const float* W2    = (const float*)d_in[5];   // [8,3072,768]
    const float* B2    = (const float*)d_in[6];   // [8,768]
    float* out = (float*)d_out;

    // workspace layout
    char* ws = (char*)d_ws;
    size_t off = 0;
    int* counts  = (int*)(ws + off);  off += 64;
    int* offsets = (int*)(ws + off);  off += 64;
    int* route_e    = (int*)(ws + off); off += (size_t)T_TOK * 2 * 4;
    int* route_slot = (int*)(ws + off); off += (size_t)T_TOK * 2 * 4;
    int* tok_idx    = (int*)(ws + off); off += (size_t)EXPERTS * T_TOK * 4;
    float* tok_w    = (float*)(ws + off); off += (size_t)EXPERTS * T_TOK * 4;
    off = (off + 255) & ~(size_t)255;
    float* Hbuf = (float*)(ws + off); off += (size_t)2 * T_TOK * D_FF * 4;
    float* Ybuf = (float*)(ws + off); off += (size_t)2 * T_TOK * N_EMBD * 4;
    (void)ws_size; (void)in_sizes; (void)n_in; (void)out_size;

    moe_zero_counts<<<1, 32, 0, stream>>>(counts);
    moe_router<<<T_TOK / 8, 256, 0, stream>>>(X, NOISE, RW, counts, tok_idx,
                                              tok_w, route_e, route_slot);
    moe_scan<<<1, 32, 0, stream>>>(counts, offsets);
    {
        dim3 grid(D_FF / NT, T_TOK / 64, EXPERTS);
        moe_ffn1<<<grid, 256, 0, stream>>>(X, W1, B1, counts, offsets, tok_idx, Hbuf);
    }
    {
        dim3 grid(N_EMBD / NT, T_TOK / 64, EXPERTS);
        moe_ffn2<<<grid, 256, 0, stream>>>(Hbuf, W2, B2, counts, offsets, tok_w, Ybuf);
    }
    moe_combine<<<T_TOK, N_EMBD / 4, 0, stream>>>(Ybuf, offsets, route_e,
                                                  route_slot, out);
}